// Decoder_75084618269499
// MI455X (gfx1250) — compile-verified
//
#include <hip/hip_runtime.h>
#include <math.h>

// Problem constants (fixed by setup_inputs)
#define BB 32
#define NN 500
#define GG 500
#define HH 128
#define NHEADS 8
#define DD 16
#define KNB 16

typedef __attribute__((ext_vector_type(2))) float v2f;
typedef __attribute__((ext_vector_type(8))) float v8f;

// ---------------------------------------------------------------------------
// 1) graph mean + q_graph = mean(emb) @ Wq_graph^T      (one block per batch)
// ---------------------------------------------------------------------------
__global__ void mean_qgraph_kernel(const float* __restrict__ EMB,
                                   const float* __restrict__ Wqg,
                                   float* __restrict__ QG) {
  __shared__ float gsm[HH];
  int b = blockIdx.x, h = threadIdx.x;
  const float* e = EMB + (size_t)b * NN * HH;
  float s = 0.f;
  for (int n = 0; n < NN; ++n) s += e[(size_t)n * HH + h];
  gsm[h] = s * (1.0f / (float)NN);
  __syncthreads();
  const float* wr = Wqg + (size_t)h * HH;
  float q = 0.f;
  for (int i = 0; i < HH; ++i) q += gsm[i] * wr[i];
  QG[b * HH + h] = q;
}

// ---------------------------------------------------------------------------
// 2) elementwise Wsum = Wq_first + Wq_last
// ---------------------------------------------------------------------------
__global__ void add_mats_kernel(const float* __restrict__ A,
                                const float* __restrict__ B,
                                float* __restrict__ C, int n) {
  int i = blockIdx.x * blockDim.x + threadIdx.x;
  if (i < n) C[i] = A[i] + B[i];
}

// ---------------------------------------------------------------------------
// 3) gather last-node embeddings: LE[b*G+g,:] = EMB[b, last[b,g], :]
// ---------------------------------------------------------------------------
__global__ void gather_rows_kernel(const float* __restrict__ EMB,
                                   const int* __restrict__ LASTN,
                                   float* __restrict__ LE) {
  int i = blockIdx.x * blockDim.x + threadIdx.x;   // over B*G*H
  if (i >= BB * GG * HH) return;
  int bg = i >> 7, h = i & (HH - 1);
  int b = bg / GG;
  int ln = LASTN[bg];
  LE[i] = EMB[(((size_t)b * NN) + ln) * HH + h];
}

// ---------------------------------------------------------------------------
// 4) generic WMMA f32 GEMM:  OUT[M x 128] = X[M x 128] @ W^T (+ optional bias)
//    one wave per 16x16 output tile, V_WMMA_F32_16X16X4_F32 over K=128.
//    blockDim = 128 (4 waves, 4 tiles per block). M must be multiple of 16.
// ---------------------------------------------------------------------------
__global__ void gemm_xwT_kernel(const float* __restrict__ X,
                                const float* __restrict__ W,
                                const float* __restrict__ bias,
                                float* __restrict__ OUT, int M) {
  const int NT = HH / 16;                       // 8 tiles across output cols
  int tile = blockIdx.x * (blockDim.x >> 5) + (threadIdx.x >> 5);
  int tm = tile / NT, tn = tile % NT;
  int lane = threadIdx.x & 31;
  int half = lane >> 4, ml = lane & 15;
  const float* xrow = X + (size_t)(tm * 16 + ml) * HH;
  const float* wrow = W + (size_t)(tn * 16 + ml) * HH;
  v8f c = {};
#pragma unroll 4
  for (int k0 = 0; k0 < HH; k0 += 4) {
    int k = k0 + 2 * half;
    v2f a = *(const v2f*)(xrow + k);
    v2f b = *(const v2f*)(wrow + k);
    c = __builtin_amdgcn_wmma_f32_16x16x4_f32(false, a, false, b,
                                              (short)0, c, false, false);
  }
  float bc = bias ? bias[tn * 16 + ml] : 0.f;
#pragma unroll
  for (int r = 0; r < 8; ++r) {
    int row = tm * 16 + r + 8 * half;
    OUT[(size_t)row * HH + tn * 16 + ml] = c[r] + bc;
  }
}

// ---------------------------------------------------------------------------
// 5) QS[b,g,:] += QG[b,:]   (broadcast q_graph over group dim)
// ---------------------------------------------------------------------------
__global__ void add_qgraph_kernel(float* __restrict__ QS,
                                  const float* __restrict__ QG) {
  int i = blockIdx.x * blockDim.x + threadIdx.x;   // over B*G*H
  if (i >= BB * GG * HH) return;
  int h = i & (HH - 1);
  int b = i / (GG * HH);
  QS[i] += QG[b * HH + h];
}

// ---------------------------------------------------------------------------
// 6) top-16 nearest neighbors per (b,g); stable ties toward lower index,
//    masked (-inf) nodes pushed to +inf distance.  One thread per row.
// ---------------------------------------------------------------------------
__global__ void topk_neighbors_kernel(const float* __restrict__ COORD,
                                      const int* __restrict__ LASTN,
                                      const float* __restrict__ MASK,
                                      int* __restrict__ NBR) {
  int idx = blockIdx.x * blockDim.x + threadIdx.x;   // over B*G
  if (idx >= BB * GG) return;
  int b = idx / GG;
  int ln = LASTN[idx];
  const float* cb = COORD + (size_t)b * NN * 2;
  float lx = cb[ln * 2 + 0], ly = cb[ln * 2 + 1];
  const float* mrow = MASK + (size_t)idx * NN;
  float kd[KNB];
  int ki[KNB];
#pragma unroll
  for (int j = 0; j < KNB; ++j) { kd[j] = INFINITY; ki[j] = 0; }
  for (int n = 0; n < NN; ++n) {
    float dx = lx - cb[n * 2], dy = ly - cb[n * 2 + 1];
    float dsq = dx * dx + dy * dy;           // monotonic with sqrt distance
    if (mrow[n] == -INFINITY) dsq = INFINITY;
    if (dsq < kd[KNB - 1]) {
      int j = KNB - 1;
      while (j > 0 && kd[j - 1] > dsq) { kd[j] = kd[j - 1]; ki[j] = ki[j - 1]; --j; }
      kd[j] = dsq; ki[j] = n;
    }
  }
#pragma unroll
  for (int j = 0; j < KNB; ++j) NBR[idx * KNB + j] = ki[j];
}

// ---------------------------------------------------------------------------
// 7) sparse 8-head attention over 16 gathered neighbors. Block per (b,g).
// ---------------------------------------------------------------------------
__global__ void sparse_attention_kernel(const float* __restrict__ QS,
                                        const float* __restrict__ KK,
                                        const float* __restrict__ VV,
                                        const int* __restrict__ NBR,
                                        float* __restrict__ AOUT) {
  __shared__ float Ksm[KNB][HH];
  __shared__ float Vsm[KNB][HH];
  __shared__ float qsm[HH];
  __shared__ float sc[HH];                  // 8 heads * 16 neighbors
  int bg = blockIdx.x;
  int b = bg / GG;
  int tid = threadIdx.x;                    // 128 threads
  qsm[tid] = QS[(size_t)bg * HH + tid];
#pragma unroll
  for (int j = 0; j < KNB; ++j) {
    int nb = NBR[bg * KNB + j];
    size_t row = (((size_t)b * NN) + nb) * HH;
    Ksm[j][tid] = KK[row + tid];
    Vsm[j][tid] = VV[row + tid];
  }
  __syncthreads();
  int h = tid >> 4, j = tid & 15;
  float s = 0.f;
#pragma unroll
  for (int t = 0; t < DD; ++t) s += qsm[h * DD + t] * Ksm[j][h * DD + t];
  sc[tid] = s * 0.25f;                      // 1/sqrt(d), d=16
  __syncthreads();
  float m = -INFINITY;
#pragma unroll
  for (int t = 0; t < KNB; ++t) m = fmaxf(m, sc[h * KNB + t]);
  float sum = 0.f;
#pragma unroll
  for (int t = 0; t < KNB; ++t) sum += expf(sc[h * KNB + t] - m);
  float w = expf(sc[tid] - m) / sum;
  __syncthreads();
  sc[tid] = w;
  __syncthreads();
  float o = 0.f;
#pragma unroll
  for (int jj = 0; jj < KNB; ++jj) o += sc[h * KNB + jj] * Vsm[jj][h * DD + j];
  AOUT[(size_t)bg * HH + h * DD + j] = o;
}

// ---------------------------------------------------------------------------
// 8) pointer scores + tanh clip + mask + row softmax, fused, WMMA f32.
//    One block handles batch b, 16 query rows, all 512 padded columns.
//    grid = B * 32 blocks of 256 threads (8 waves x 4 column tiles each).
// ---------------------------------------------------------------------------
__global__ void pointer_score_softmax_kernel(const float* __restrict__ FQ,
                                             const float* __restrict__ EMB,
                                             const float* __restrict__ MASK,
                                             float* __restrict__ OUT) {
  __shared__ float sm[16 * 512];
  __shared__ float rmax[16];
  __shared__ float rsum[16];
  int b = blockIdx.x >> 5;
  int tm = blockIdx.x & 31;
  int tid = threadIdx.x;
  int wave = tid >> 5, lane = tid & 31;
  int half = lane >> 4, ml = lane & 15;
  int g0 = tm * 16;
  const float* eb = EMB + (size_t)b * NN * HH;
  int ar = g0 + ml; if (ar > NN - 1) ar = NN - 1;     // clamp OOB query rows
  const float* xp = FQ + (((size_t)b * GG) + ar) * HH;
#pragma unroll
  for (int i = 0; i < 4; ++i) {
    int tn = wave * 4 + i;
    int n0 = tn * 16;
    int br = n0 + ml; if (br > NN - 1) br = NN - 1;   // clamp OOB key rows
    const float* bp = eb + (size_t)br * HH;
    v8f c = {};
#pragma unroll 4
    for (int k0 = 0; k0 < HH; k0 += 4) {
      int k = k0 + 2 * half;
      v2f a = *(const v2f*)(xp + k);
      v2f bv = *(const v2f*)(bp + k);
      c = __builtin_amdgcn_wmma_f32_16x16x4_f32(false, a, false, bv,
                                                (short)0, c, false, false);
    }
#pragma unroll
    for (int r = 0; r < 8; ++r) {
      int lr = r + 8 * half;           // local row 0..15
      int g = g0 + lr;
      int n = n0 + ml;
      float v;
      if (n < NN) {
        float msk = (g < GG) ? MASK[((size_t)b * GG + g) * NN + n] : 0.f;
        v = tanhf(c[r]) * 10.0f + msk;
      } else {
        v = -INFINITY;
      }
      sm[lr * 512 + n] = v;
    }
  }
  __syncthreads();
  if (tid < 16) {
    float m = -INFINITY;
    for (int n = 0; n < NN; ++n) m = fmaxf(m, sm[tid * 512 + n]);
    float s = 0.f;
    for (int n = 0; n < NN; ++n) s += expf(sm[tid * 512 + n] - m);
    rmax[tid] = m; rsum[tid] = s;
  }
  __syncthreads();
  for (int idx = tid; idx < 16 * NN; idx += 256) {
    int r = idx / NN, n = idx - r * NN;
    int g = g0 + r;
    if (g < GG)
      OUT[((size_t)b * GG + g) * NN + n] = expf(sm[r * 512 + n] - rmax[r]) / rsum[r];
  }
}

// ---------------------------------------------------------------------------
// launcher
// ---------------------------------------------------------------------------
extern "C" void kernel_launch(void* const* d_in, const int* in_sizes, int n_in,
                              void* d_out, int out_size, void* d_ws, size_t ws_size,
                              hipStream_t stream) {
  (void)in_sizes; (void)n_in; (void)out_size; (void)ws_size;

  const float* coords = (const float*)d_in[0];   // (B,N,2)
  const float* emb    = (const float*)d_in[1];   // (B,N,H)
  const int*   lastn  = (const int*)  d_in[2];   // (B,G)
  const float* mask   = (const float*)d_in[3];   // (B,G,N)
  const float* Wqg    = (const float*)d_in[4];
  const float* Wqf    = (const float*)d_in[5];
  const float* Wql    = (const float*)d_in[6];
  const float* Wk     = (const float*)d_in[7];
  const float* Wv     = (const float*)d_in[8];
  const float* Wc     = (const float*)d_in[9];
  const float* bc     = (const float*)d_in[10];
  float* out = (float*)d_out;                    // (B,G,N)

  // workspace layout (floats), with aliasing: bufA = last_emb -> attn_out,
  // bufB = q_sum -> final_q
  const size_t ROWS = (size_t)BB * GG;           // 16000 (== B*N too)
  float* ws   = (float*)d_ws;
  float* qg   = ws;                              // B*H
  float* wsum = qg + (size_t)BB * HH;            // H*H
  float* bufA = wsum + (size_t)HH * HH;          // ROWS*H
  float* KK   = bufA + ROWS * HH;                // ROWS*H
  float* VV   = KK + ROWS * HH;                  // ROWS*H
  float* bufB = VV + ROWS * HH;                  // ROWS*H
  int*   NBR  = (int*)(bufB + ROWS * HH);        // ROWS*16

  const int elemsBGH = BB * GG * HH;             // 2,048,000
  const int gemmBlocks = (int)((ROWS / 16) * (HH / 16) / 4);  // 2000

  // 1) graph mean + q_graph
  mean_qgraph_kernel<<<BB, HH, 0, stream>>>(emb, Wqg, qg);
  // 2) Wsum = Wq_first + Wq_last
  add_mats_kernel<<<(HH * HH + 255) / 256, 256, 0, stream>>>(Wqf, Wql, wsum, HH * HH);
  // 3) gather last-node embeddings
  gather_rows_kernel<<<(elemsBGH + 255) / 256, 256, 0, stream>>>(emb, lastn, bufA);
  // 4) K / V projections
  gemm_xwT_kernel<<<gemmBlocks, 128, 0, stream>>>(emb, Wk, nullptr, KK, (int)ROWS);
  gemm_xwT_kernel<<<gemmBlocks, 128, 0, stream>>>(emb, Wv, nullptr, VV, (int)ROWS);
  // 5) q_sum = last_emb @ (Wq_first+Wq_last)^T ; then += q_graph
  gemm_xwT_kernel<<<gemmBlocks, 128, 0, stream>>>(bufA, wsum, nullptr, bufB, (int)ROWS);
  add_qgraph_kernel<<<(elemsBGH + 255) / 256, 256, 0, stream>>>(bufB, qg);
  // 6) top-16 neighbor selection
  topk_neighbors_kernel<<<(BB * GG + 255) / 256, 256, 0, stream>>>(coords, lastn, mask, NBR);
  // 7) sparse multi-head attention -> attn_out (bufA)
  sparse_attention_kernel<<<BB * GG, HH, 0, stream>>>(bufB, KK, VV, NBR, bufA);
  // 8) final_q = attn_out @ Wc^T + bc  (bufB)
  gemm_xwT_kernel<<<gemmBlocks, 128, 0, stream>>>(bufA, Wc, bc, bufB, (int)ROWS);
  // 9) fused pointer scores + tanh clip + mask + softmax -> probs
  pointer_score_softmax_kernel<<<BB * 32, 256, 0, stream>>>(bufB, emb, mask, out);
}